// SparseProductNodes_27608049779346
// MI455X (gfx1250) — compile-verified
//
#include <hip/hip_runtime.h>
#include <hip/hip_bf16.h>

// -----------------------------------------------------------------------------
// Sparse product-node forward: out[i] = sum_j child[flat_idx[j]] for segs[j]==i
// Pure data movement (256 MB index streams, L2-resident 8 MB gather table).
// CDNA5 path: TDM tensor_load_to_lds double-buffered streaming of the two
// index arrays into LDS (TENSORcnt-synchronized), ds_load_b128 LDS reads,
// batched 16-wide gather clause, sorted-segment run accumulation -> few f32
// global atomics.
// -----------------------------------------------------------------------------

#define CHUNK   4096          // edges per chunk  (16 KB per stream per buffer)
#define EPT     16            // edges per thread (CHUNK / 256)
#define NBLKMAX 2048          // persistent workgroups

typedef unsigned int u32x4 __attribute__((ext_vector_type(4)));
typedef int          i32x4 __attribute__((ext_vector_type(4)));
typedef int          i32x8 __attribute__((ext_vector_type(8)));

// 1D TDM copy: global -> LDS, `valid` elements defined (OOB -> zeros),
// tile is CHUNK u32 elements. Descriptor per CDNA5 ISA ch.8 (D# groups).
__device__ __forceinline__ void tdm_copy_1d(unsigned lds_byte_addr,
                                            const void* gptr,
                                            unsigned valid_elems) {
  unsigned long long ga = (unsigned long long)(uintptr_t)gptr;

  u32x4 g0;
  g0.x = 1u;                                           // count=1, user descriptor
  g0.y = lds_byte_addr;                                // lds_addr (bytes)
  g0.z = (unsigned)(ga & 0xffffffffull);               // global_addr[31:0]
  g0.w = (unsigned)((ga >> 32) & 0x1ffffffull)         // global_addr[56:32]
       | (2u << 30);                                   // type = 2 ("image")

  i32x8 g1 = {0, 0, 0, 0, 0, 0, 0, 0};
  g1[0] = (int)(2u << 16);                             // wg_mask=0, data_size=2 (4B)
  g1[1] = (int)((valid_elems & 0xffffu) << 16);        // tensor_dim0[15:0]  -> bits 63:48
  g1[2] = (int)(valid_elems >> 16);                    // tensor_dim0[31:16] -> bits 79:64
  g1[3] = (int)((unsigned)CHUNK << 16);                // tile_dim0 -> bits 127:112; tensor_dim1=0
  // g1[4..7] = 0 : tile_dim1/2 unused (1D), strides unused

  i32x4 z4 = {0, 0, 0, 0};
#if __clang_major__ >= 23
  i32x8 z8 = {0, 0, 0, 0, 0, 0, 0, 0};
  __builtin_amdgcn_tensor_load_to_lds(g0, g1, z4, z4, z8, 0);
#else
  __builtin_amdgcn_tensor_load_to_lds(g0, g1, z4, z4, 0);
#endif
}

__global__ void zero_f32_kernel(float* __restrict__ p, int n) {
  int i = blockIdx.x * blockDim.x + threadIdx.x;
  if (i < n) p[i] = 0.0f;
}

__global__ void __launch_bounds__(256)
spn_segsum_kernel(const float* __restrict__ child,
                  const unsigned* __restrict__ fidx,
                  const unsigned* __restrict__ segs,
                  float* __restrict__ out,
                  unsigned E, unsigned nChunks) {
  __shared__ unsigned lidx[2][CHUNK];
  __shared__ unsigned lseg[2][CHUNK];

  const unsigned t = threadIdx.x;
  // readfirstlane -> compiler-known uniform -> scalar branch around TDM issue
  const unsigned waveid = __builtin_amdgcn_readfirstlane(t >> 5);

  unsigned c = blockIdx.x;
  if (c >= nChunks) return;

  // Prime the pipeline: DMA first chunk into buffer 0 (wave 0 only).
  if (waveid == 0) {
    unsigned base  = c * CHUNK;
    unsigned valid = E - base; if (valid > CHUNK) valid = CHUNK;
    tdm_copy_1d((unsigned)(uintptr_t)&lidx[0][0], fidx + base, valid);
    tdm_copy_1d((unsigned)(uintptr_t)&lseg[0][0], segs + base, valid);
  }

  int buf = 0;
  for (; c < nChunks; c += gridDim.x) {
    const unsigned cn = c + gridDim.x;
    const bool hasNext = (cn < nChunks);

    // Overlap: issue next chunk's DMA pair into the other buffer.
    if (hasNext && waveid == 0) {
      unsigned nbase = cn * CHUNK;
      unsigned valid = E - nbase; if (valid > CHUNK) valid = CHUNK;
      tdm_copy_1d((unsigned)(uintptr_t)&lidx[buf ^ 1][0], fidx + nbase, valid);
      tdm_copy_1d((unsigned)(uintptr_t)&lseg[buf ^ 1][0], segs + nbase, valid);
    }

    // TDM completes in order per wave: <=2 outstanding means the current
    // chunk's pair has landed (the 2 newest are the next chunk's pair).
    if (hasNext) __builtin_amdgcn_s_wait_tensorcnt(2);
    else         __builtin_amdgcn_s_wait_tensorcnt(0);
    __syncthreads();   // publish DMA'd LDS to all 8 waves

    // ---- compute: 16 consecutive edges per thread --------------------------
    const unsigned base = c * CHUNK;
    const unsigned j0   = base + t * EPT;
    int kmax = 0;
    if (j0 < E) {
      unsigned rem = E - j0;
      kmax = (rem > (unsigned)EPT) ? EPT : (int)rem;
    }

    // Wide LDS reads: 4x ds_load_b128 per stream (64-B aligned per thread).
    const uint4* li4 = (const uint4*)&lidx[buf][t * EPT];
    const uint4* ls4 = (const uint4*)&lseg[buf][t * EPT];
    unsigned vi[EPT], si[EPT];
    #pragma unroll
    for (int q = 0; q < EPT / 4; ++q) {
      uint4 a = li4[q];
      uint4 b = ls4[q];
      vi[q * 4 + 0] = a.x; vi[q * 4 + 1] = a.y; vi[q * 4 + 2] = a.z; vi[q * 4 + 3] = a.w;
      si[q * 4 + 0] = b.x; si[q * 4 + 1] = b.y; si[q * 4 + 2] = b.z; si[q * 4 + 3] = b.w;
    }

    // Batched gather clause: 16 outstanding global_load_b32 into the
    // L2-resident value table (speculative lanes hit child[0], always legal).
    float v[EPT];
    #pragma unroll
    for (int k = 0; k < EPT; ++k) v[k] = child[vi[k]];

    // Register-only run accumulation over sorted segment ids.
    float acc = 0.0f;
    int   cur = -1;
    #pragma unroll
    for (int k = 0; k < EPT; ++k) {
      if (k < kmax) {
        int s = (int)si[k];
        if (s != cur) {
          if (cur >= 0) atomicAdd(out + cur, acc);
          cur = s;
          acc = 0.0f;
        }
        acc += v[k];
      }
    }
    if (cur >= 0) atomicAdd(out + cur, acc);
    // ------------------------------------------------------------------------

    __syncthreads();   // all waves done reading `buf` before it is re-DMA'd
    buf ^= 1;
  }
}

extern "C" void kernel_launch(void* const* d_in, const int* in_sizes, int n_in,
                              void* d_out, int out_size, void* d_ws, size_t ws_size,
                              hipStream_t stream) {
  const float*    child = (const float*)d_in[0];
  const unsigned* fidx  = (const unsigned*)d_in[1];
  const unsigned* segs  = (const unsigned*)d_in[2];
  // d_in[3] = num_nodes scalar on device; out_size already equals N (batch=1).
  float* out = (float*)d_out;

  const unsigned E = (unsigned)in_sizes[1];
  const int      N = out_size;

  // Atomic accumulation requires a clean output every call.
  zero_f32_kernel<<<(N + 255) / 256, 256, 0, stream>>>(out, N);

  const unsigned nChunks = (E + CHUNK - 1) / CHUNK;
  unsigned nb = nChunks < NBLKMAX ? nChunks : NBLKMAX;
  if (nb == 0) return;
  spn_segsum_kernel<<<nb, 256, 0, stream>>>(child, fidx, segs, out, E, nChunks);
}